// SelfSimilarityProbDistance_48215302865053
// MI455X (gfx1250) — compile-verified
//
#include <hip/hip_runtime.h>
#include <hip/hip_bf16.h>
#include <float.h>

typedef float v2f __attribute__((ext_vector_type(2)));
typedef float v8f __attribute__((ext_vector_type(8)));

__device__ __constant__ float kInvTemp = 1.0f / 13.544f;
#define K_LOG2E 1.4426950408889634f
#define K_LN2   0.6931471805599453f
#define K_EPS   1e-8f

// ---------------------------------------------------------------------------
// K1: raw tsm tile (16x16 i x j per block) + masked per-block partial sums.
// Symmetry: tsm[i,j] == tsm[j,i], so only upper-triangular tiles (ti <= tj)
// are computed; off-diagonal tiles are mirrored via an LDS transpose and
// contribute x2 to the partial sums. X tiles staged via async-to-LDS.
// ---------------------------------------------------------------------------
__global__ __launch_bounds__(256) void k_tsm(
    const float* __restrict__ X, const unsigned char* __restrict__ mask,
    const float* __restrict__ pa, const float* __restrict__ pb,
    float* __restrict__ tsm, float* __restrict__ partials,
    int N, int tilesPerRow, int tilesPerBatch)
{
  __shared__ __align__(16) float sXi[16 * 16];
  __shared__ __align__(16) float sXj[16 * 16];
  __shared__ float sT[16 * 17];       // padded transpose buffer (bank-safe)
  __shared__ float r1[256];
  __shared__ float r2[256];

  const int tid = threadIdx.x;
  const int bid = blockIdx.x;
  const int b   = bid / tilesPerBatch;
  const int t   = bid - b * tilesPerBatch;
  const int ti  = t / tilesPerRow;
  const int tj  = t - ti * tilesPerRow;

  if (ti > tj) {                      // mirrored by the (tj,ti) block
    if (tid == 0) { partials[2 * bid + 0] = 0.0f; partials[2 * bid + 1] = 0.0f; }
    return;
  }
  const int i0 = ti * 16;
  const int j0 = tj * 16;

  const float* gXi = X + ((size_t)b * N + i0) * 16;   // 16 rows x 16 f32, contiguous 1KB
  const float* gXj = X + ((size_t)b * N + j0) * 16;

  // Stage both tiles: 128 lanes x 16B via async-to-LDS (ASYNCcnt path).
  if (tid < 64) {
    unsigned lds = (unsigned)(size_t)(&sXi[tid * 4]);
    unsigned long long ga = (unsigned long long)(size_t)gXi + (unsigned long long)(tid * 16);
    asm volatile("global_load_async_to_lds_b128 %0, %1, off" :: "v"(lds), "v"(ga) : "memory");
  } else if (tid < 128) {
    const int u = tid - 64;
    unsigned lds = (unsigned)(size_t)(&sXj[u * 4]);
    unsigned long long ga = (unsigned long long)(size_t)gXj + (unsigned long long)(u * 16);
    asm volatile("global_load_async_to_lds_b128 %0, %1, off" :: "v"(lds), "v"(ga) : "memory");
  }
  if (tid < 128) {
#if __has_builtin(__builtin_amdgcn_s_wait_asynccnt)
    __builtin_amdgcn_s_wait_asynccnt(0);
#else
    asm volatile("s_wait_asynccnt 0x0" ::: "memory");
#endif
  }
  __syncthreads();

  const float sa  = pa[0];
  const float sbv = pb[0];
  const float c1  = sa * K_LOG2E;    // exp(-u) = exp2(c1*d2 + c0)
  const float c0  = -sbv * K_LOG2E;

  const int il = tid >> 4;           // local i row
  const int jl = tid & 15;           // local j row
  float xi[16], xj[16];
#pragma unroll
  for (int k = 0; k < 4; ++k) {
    float4 a = ((const float4*)sXi)[il * 4 + k];
    xi[4 * k + 0] = a.x; xi[4 * k + 1] = a.y; xi[4 * k + 2] = a.z; xi[4 * k + 3] = a.w;
    float4 c = ((const float4*)sXj)[jl * 4 + k];
    xj[4 * k + 0] = c.x; xj[4 * k + 1] = c.y; xj[4 * k + 2] = c.z; xj[4 * k + 3] = c.w;
  }

  float acc = 0.0f;
#pragma unroll
  for (int p = 0; p < 16; ++p) {
    const float xp = xi[p];
#pragma unroll
    for (int q = 0; q < 16; ++q) {
      float d = xp - xj[q];
      float m = d * d;
      float e = __builtin_amdgcn_exp2f(fmaf(m, c1, c0));       // exp(a*d2 - b)
      float r = __builtin_amdgcn_rcpf(1.0f + e);               // sigmoid
      float l = __builtin_amdgcn_logf(r + K_EPS);              // log2(sig+eps)
      acc += l;
    }
  }
  const float tv = acc * (-K_LN2 / 256.0f);  // mean of -ln(sig+eps)

  const int gi = i0 + il, gj = j0 + jl;
  tsm[((size_t)b * N + gi) * N + gj] = tv;

  // Mirror off-diagonal tile with a coalesced transposed store.
  if (ti != tj) {
    sT[il * 17 + jl] = tv;
    __syncthreads();
    const float tvT = sT[jl * 17 + il];                // = tv(jl, il)
    tsm[((size_t)b * N + j0 + il) * N + (i0 + jl)] = tvT;
  }

  const float w = (mask[(size_t)b * N + gi] && mask[(size_t)b * N + gj]) ? 1.0f : 0.0f;
  const float scale = (ti == tj) ? 1.0f : 2.0f;        // mirrored entries count twice
  const float v = tv * w;
  r1[tid] = v * scale;
  r2[tid] = v * v * scale;
  __syncthreads();
  for (int s = 128; s > 0; s >>= 1) {
    if (tid < s) { r1[tid] += r1[tid + s]; r2[tid] += r2[tid + s]; }
    __syncthreads();
  }
  if (tid == 0) {
    partials[2 * bid + 0] = r1[0];
    partials[2 * bid + 1] = r2[0];
  }
}

// ---------------------------------------------------------------------------
// K2: one wave per batch. Sum the per-tile partials with WMMA (A = data,
// B = all-ones -> D rows are exact f32 row-sums; layout-independent, exact,
// deterministic). Then mean / 1/std from sum & sumsq + mask count.
// ---------------------------------------------------------------------------
__device__ inline v8f wmma_rowsum_acc(v2f a, v2f ones, v8f acc) {
#if __has_builtin(__builtin_amdgcn_wmma_f32_16x16x4_f32)
  return __builtin_amdgcn_wmma_f32_16x16x4_f32(false, a, false, ones,
                                               (short)0, acc, false, false);
#else
  asm volatile("v_wmma_f32_16x16x4_f32 %0, %1, %2, %0\n\ts_nop 5"
               : "+v"(acc) : "v"(a), "v"(ones));
  return acc;
#endif
}

__global__ __launch_bounds__(32) void k_stats(
    const float* __restrict__ partials, const unsigned char* __restrict__ mask,
    float* __restrict__ stats, int N, int P)
{
  const int b = blockIdx.x;
  const int lane = threadIdx.x;
  const float* pt = partials + (size_t)2 * b * P;

  v8f acc1 = {0, 0, 0, 0, 0, 0, 0, 0};
  v8f acc2 = {0, 0, 0, 0, 0, 0, 0, 0};
  v2f ones; ones.x = 1.0f; ones.y = 1.0f;

  const int chunks = P / 64;   // 64 tiles (128 floats) per WMMA pair
  for (int c = 0; c < chunks; ++c) {
    float4 d4 = ((const float4*)pt)[c * 32 + lane];  // {sum0, sq0, sum1, sq1}
    v2f a1; a1.x = d4.x; a1.y = d4.z;
    v2f a2; a2.x = d4.y; a2.y = d4.w;
    acc1 = wmma_rowsum_acc(a1, ones, acc1);
    acc2 = wmma_rowsum_acc(a2, ones, acc2);
  }
  float s1 = acc1[0] + acc1[1] + acc1[2] + acc1[3] + acc1[4] + acc1[5] + acc1[6] + acc1[7];
  float s2 = acc2[0] + acc2[1] + acc2[2] + acc2[3] + acc2[4] + acc2[5] + acc2[6] + acc2[7];
  s1 += __shfl_xor(s1, 16);   // lanes 0..15 held M=0..7, 16..31 held M=8..15
  s2 += __shfl_xor(s2, 16);

  int cnt = 0;
  for (int k = lane; k < N; k += 32) cnt += mask[(size_t)b * N + k] ? 1 : 0;
  for (int o = 16; o; o >>= 1) cnt += __shfl_xor(cnt, o);

  if (lane == 0) {
    for (int k = chunks * 64; k < P; ++k) { s1 += pt[2 * k]; s2 += pt[2 * k + 1]; }
    const float n = (float)cnt * (float)cnt;
    const float mean = s1 / n;
    const float var = (s2 - s1 * s1 / n) / (n - 1.0f);  // unbiased
    stats[2 * b + 0] = mean;
    stats[2 * b + 1] = 1.0f / sqrtf(var);
  }
}

// ---------------------------------------------------------------------------
// K3: per-row masked softmax of -(t-mean)/std / T, in place over d_out.
// ---------------------------------------------------------------------------
__global__ __launch_bounds__(256) void k_softmax(
    float* __restrict__ tsm, const unsigned char* __restrict__ mask,
    const float* __restrict__ stats, int N)
{
  __shared__ float sv[2048];
  __shared__ float red[256];
  const int tid = threadIdx.x;
  const int rid = blockIdx.x;
  const int b = rid / N;
  const int i = rid - b * N;
  float* row = tsm + (size_t)rid * N;
  const unsigned char* mrow = mask + (size_t)b * N;

  if (!mrow[i]) {                      // whole row masked -> zeros
    for (int j = tid; j < N; j += 256) row[j] = 0.0f;
    return;
  }
  const float mean = stats[2 * b + 0];
  const float rinv = stats[2 * b + 1];

  for (int j = tid; j < N; j += 256) {
    float tv = row[j];
    sv[j] = mrow[j] ? (-(tv - mean) * rinv * kInvTemp) : -FLT_MAX;
  }
  __syncthreads();

  float m = -FLT_MAX;
  for (int j = tid; j < N; j += 256) m = fmaxf(m, sv[j]);
  red[tid] = m;
  __syncthreads();
  for (int s = 128; s > 0; s >>= 1) {
    if (tid < s) red[tid] = fmaxf(red[tid], red[tid + s]);
    __syncthreads();
  }
  const float M = red[0];
  __syncthreads();

  float sm = 0.0f;
  for (int j = tid; j < N; j += 256) {
    float e = __builtin_amdgcn_exp2f((sv[j] - M) * K_LOG2E);
    sv[j] = e;
    sm += e;
  }
  red[tid] = sm;
  __syncthreads();
  for (int s = 128; s > 0; s >>= 1) {
    if (tid < s) red[tid] += red[tid + s];
    __syncthreads();
  }
  const float rs = 1.0f / red[0];
  for (int j = tid; j < N; j += 256) row[j] = mrow[j] ? sv[j] * rs : 0.0f;
}

// ---------------------------------------------------------------------------
extern "C" void kernel_launch(void* const* d_in, const int* in_sizes, int n_in,
                              void* d_out, int out_size, void* d_ws, size_t ws_size,
                              hipStream_t stream) {
  const float* X            = (const float*)d_in[0];          // [B,N,16] f32
  const unsigned char* mask = (const unsigned char*)d_in[1];  // [B,N] bool
  const float* pa           = (const float*)d_in[2];          // sigmoid_a (precomputed)
  const float* pbv          = (const float*)d_in[3];          // sigmoid_b
  float* out = (float*)d_out;                                 // [B,N,N] f32

  const int BN  = in_sizes[1];        // B*N
  const int N   = out_size / BN;      // 512
  const int B   = BN / N;             // 2
  const int tpr = N / 16;
  const int tpb = tpr * tpr;

  float* partials = (float*)d_ws;                       // [B*tpb][2] f32
  float* stats    = partials + (size_t)2 * B * tpb;     // [B][2] f32 (mean, 1/std)

  k_tsm<<<B * tpb, 256, 0, stream>>>(X, mask, pa, pbv, out, partials, N, tpr, tpb);
  k_stats<<<B, 32, 0, stream>>>(partials, mask, stats, N, tpb);
  k_softmax<<<B * N, 256, 0, stream>>>(out, mask, stats, N);
}